// Detect_75033078661307
// MI455X (gfx1250) — compile-verified
//
#include <hip/hip_runtime.h>
#include <hip/hip_bf16.h>
#include <stdint.h>

#define B_      32
#define N_      8732
#define C_      81
#define TOPK    200
#define CONF_T  0.01f
#define NMS_T   0.45f
#define NEG_INF (-__builtin_inff())

typedef __attribute__((ext_vector_type(2))) float v2f;
typedef __attribute__((ext_vector_type(8))) float v8f;

// ---- gfx1250 async gather: one 4B element per lane, global -> LDS (ASYNCcnt) ----
__device__ __forceinline__ void async_gather_b32(uint32_t lds_off, const float* gptr) {
    asm volatile("global_load_async_to_lds_b32 %0, %1, off"
                 :: "v"(lds_off), "v"((uint64_t)(uintptr_t)gptr)
                 : "memory");
}
__device__ __forceinline__ void wait_async0() {
    asm volatile("s_wait_asynccnt 0" ::: "memory");
}

// ============ Kernel A: per-anchor softmax stats (online) + box decode ============
__global__ __launch_bounds__(256)
void softmax_decode_kernel(const float* __restrict__ loc,
                           const float* __restrict__ conf,
                           const float* __restrict__ dbox,
                           float2* __restrict__ denom,
                           float4* __restrict__ boxes) {
    int idx = blockIdx.x * blockDim.x + threadIdx.x;
    if (idx >= B_ * N_) return;
    int n = idx % N_;

    const float* cp = conf + (size_t)idx * C_;
    float m = cp[0], s = 1.0f;
#pragma unroll 4
    for (int c = 1; c < C_; ++c) {
        float x = cp[c];
        if (x > m) { s = s * __expf(m - x) + 1.0f; m = x; }
        else       { s += __expf(x - m); }
    }
    denom[idx] = make_float2(m, s);

    const float* lp = loc + (size_t)idx * 4;
    const float* dp = dbox + (size_t)n * 4;
    float cx = dp[0] + lp[0] * 0.1f * dp[2];
    float cy = dp[1] + lp[1] * 0.1f * dp[3];
    float w  = dp[2] * __expf(lp[2] * 0.2f);
    float h  = dp[3] * __expf(lp[3] * 0.2f);
    float x0 = cx - 0.5f * w, y0 = cy - 0.5f * h;
    boxes[idx] = make_float4(x0, y0, x0 + w, y0 + h);
}

// ---- block-wide argmax, lowest-index tie-break (wave32 shuffles + tiny LDS) ----
__device__ __forceinline__ void block_argmax(float v, int i,
                                             volatile float* wv, volatile int* wi,
                                             volatile float* selv, volatile int* seli) {
#pragma unroll
    for (int off = 16; off > 0; off >>= 1) {
        float v2 = __shfl_down(v, off, 32);
        int   i2 = __shfl_down(i, off, 32);
        if (v2 > v || (v2 == v && (unsigned)i2 < (unsigned)i)) { v = v2; i = i2; }
    }
    int wid = threadIdx.x >> 5;
    if ((threadIdx.x & 31) == 0) { wv[wid] = v; wi[wid] = i; }
    __syncthreads();
    if (threadIdx.x == 0) {
#pragma unroll
        for (int w = 1; w < 8; ++w) {
            float v2 = wv[w]; int i2 = wi[w];
            if (v2 > v || (v2 == v && (unsigned)i2 < (unsigned)i)) { v = v2; i = i2; }
        }
        *selv = v; *seli = i;
    }
    __syncthreads();
}

// ============ Kernel C: one workgroup per (batch, class>0) — topk + NMS ============
__global__ __launch_bounds__(256)
void nms_kernel(const float* __restrict__ conf,
                const float2* __restrict__ denom,
                const float4* __restrict__ boxes,
                float* __restrict__ out) {
    __shared__ float    sc[N_];            // class scores (masked)
    __shared__ float    wv[8];
    __shared__ int      wi[8];
    __shared__ float    selv_s;
    __shared__ int      seli_s;
    __shared__ float    tops[TOPK];
    __shared__ int      topi[TOPK];
    __shared__ float    cb[208][4];        // candidate corner boxes (padded)
    __shared__ float    ca[208];           // candidate areas
    __shared__ unsigned msk[208][8];       // keep-bit rows (each word single-owner)
    __shared__ unsigned aw[8];             // active bitset

    const int tid = threadIdx.x;
    const int b   = blockIdx.x / 80;
    const int cls = 1 + (blockIdx.x % 80);

    // --- Stage 1: async per-lane gather of strided column conf[b,:,cls] into LDS.
    // conf (90.5MB) is L2-resident after kernel A on the 192MB-L2 MI455X.
    {
        const float* base = conf + (size_t)b * N_ * C_ + cls;
        uint32_t sc0 = (uint32_t)(uintptr_t)(&sc[0]);
        for (int n = tid; n < N_; n += 256)
            async_gather_b32(sc0 + 4u * (uint32_t)n, base + (size_t)n * C_);
        wait_async0();
    }
    __syncthreads();

    // softmax + confidence threshold mask, in place
    for (int n = tid; n < N_; n += 256) {
        float2 ms = denom[(size_t)b * N_ + n];
        float  s  = __expf(sc[n] - ms.x) / ms.y;
        sc[n] = (s > CONF_T) ? s : NEG_INF;
    }
    __syncthreads();

    // --- Stage 2: top-200 by repeated argmax (exact, first-max tie-break)
    for (int k = tid; k < TOPK; k += 256) { tops[k] = NEG_INF; topi[k] = -1; }
    __syncthreads();
    for (int k = 0; k < TOPK; ++k) {
        float v = NEG_INF; int vi = -1;
        for (int n = tid; n < N_; n += 256) {
            float x = sc[n];
            if (x > v) { v = x; vi = n; }
        }
        block_argmax(v, vi, wv, wi, &selv_s, &seli_s);
        float mv = selv_s; int mi = seli_s;
        if (mv == NEG_INF) break;                 // uniform exit
        if (tid == 0) { tops[k] = mv; topi[k] = mi; sc[mi] = NEG_INF; }
        __syncthreads();
    }

    // --- Stage 3: gather candidate boxes + areas (zero padding to 208)
    if (tid < 208) {
        float x0 = 0, y0 = 0, x1 = 0, y1 = 0;
        if (tid < TOPK && topi[tid] >= 0) {
            float4 bb = boxes[(size_t)b * N_ + topi[tid]];
            x0 = bb.x; y0 = bb.y; x1 = bb.z; y1 = bb.w;
        }
        cb[tid][0] = x0; cb[tid][1] = y0; cb[tid][2] = x1; cb[tid][3] = y1;
        ca[tid] = (x1 - x0) * (y1 - y0);
    }
    __syncthreads();

    // --- Stage 4: IoU keep-mask. Units = (row-tile ti, mask word p) covering
    // column tiles tj0=2p, tj1=2p+1: each 32-bit mask word has exactly ONE owner,
    // so writes are plain ds_store_b32 (no atomics, no mask init). Two WMMAs per
    // unit share the A operand (area_i rank-2 update on the matrix pipe). Body is
    // straight-line (clamped dead half at p==6), so EXEC stays all-1s for WMMA.
    {
        const int  wid  = tid >> 5;
        const int  lane = tid & 31;
        const bool hi   = lane >= 16;
        const int  l15  = lane & 15;
        for (int u = wid; u < 13 * 7; u += 8) {
            int ti  = u / 7, p = u % 7;
            int tj0 = 2 * p, tj1 = 2 * p + 1;
            bool have1 = (tj1 < 13);                // uniform; false only at p==6
            int  j0  = tj0 * 16 + l15;
            int  j1c = have1 ? tj1 * 16 + l15 : l15; // clamped dead index

            float ai  = ca[ti * 16 + l15];
            float aj0 = ca[j0];
            float aj1 = ca[j1c];
            v2f a, bm0, bm1;
            a.x   = hi ? 0.0f : ai;                  // A: K=0 column = area_i
            a.y   = hi ? 0.0f : 1.0f;                // A: K=1 column = 1
            bm0.x = hi ? 0.0f : 1.0f;
            bm0.y = hi ? 0.0f : aj0;                 // B half 0: area_j (cols tj0)
            bm1.x = hi ? 0.0f : 1.0f;
            bm1.y = hi ? 0.0f : aj1;                 // B half 1: area_j (cols tj1)
            v8f c0 = {}, c1 = {};
            c0 = __builtin_amdgcn_wmma_f32_16x16x4_f32(
                     false, a, false, bm0, (short)0, c0, false, false);
            c1 = __builtin_amdgcn_wmma_f32_16x16x4_f32(
                     false, a, false, bm1, (short)0, c1, false, false);

            float ax0 = cb[j0][0],  ay0 = cb[j0][1],  ax1 = cb[j0][2],  ay1 = cb[j0][3];
            float bx0 = cb[j1c][0], by0 = cb[j1c][1], bx1 = cb[j1c][2], by1 = cb[j1c][3];
            const int i_lo = ti * 16;
#pragma unroll
            for (int r = 0; r < 8; ++r) {
                int i = i_lo + r + (hi ? 8 : 0);
                float ix0 = cb[i][0], iy0 = cb[i][1], ix1 = cb[i][2], iy1 = cb[i][3];

                float w0 = fmaxf(fminf(ix1, ax1) - fmaxf(ix0, ax0), 0.0f);
                float h0 = fmaxf(fminf(iy1, ay1) - fmaxf(iy0, ay0), 0.0f);
                float in0 = w0 * h0;
                bool  k0  = in0 <= NMS_T * fmaxf(c0[r] - in0, 1e-12f);

                float w1 = fmaxf(fminf(ix1, bx1) - fmaxf(ix0, bx0), 0.0f);
                float h1 = fmaxf(fminf(iy1, by1) - fmaxf(iy0, by0), 0.0f);
                float in1 = w1 * h1;
                bool  k1  = in1 <= NMS_T * fmaxf(c1[r] - in1, 1e-12f);

                unsigned b0 = (unsigned)__builtin_amdgcn_ballot_w32(k0);
                unsigned b1 = (unsigned)__builtin_amdgcn_ballot_w32(k1);
                b1 = have1 ? b1 : 0u;
                // row i_lo+r word: [15:0]=b0.lo16, [31:16]=b1.lo16 (lanes 0-15)
                // row i_lo+r+8:    [15:0]=b0.hi16, [31:16]=b1.hi16 (lanes 16-31)
                unsigned wA = (b0 & 0xFFFFu) | (b1 << 16);
                unsigned wB = (b0 >> 16)     | (b1 & 0xFFFF0000u);
                if ((lane & 15) == 0)
                    msk[i][p] = hi ? wB : wA;       // plain store, single owner
            }
        }
    }
    __syncthreads();

    // --- Stage 5: 200-step greedy suppression scan on LDS bitsets
    if (tid < 8) aw[tid] = 0u;
    __syncthreads();
    if (tid < TOPK && tops[tid] > CONF_T)
        atomicOr(&aw[tid >> 5], 1u << (tid & 31));
    __syncthreads();

    float* orow = out + ((size_t)b * C_ + cls) * (TOPK * 5);
    for (int step = 0; step < TOPK; ++step) {
        float v = NEG_INF; int vi = -1;
        if (tid < TOPK && ((aw[tid >> 5] >> (tid & 31)) & 1u)) { v = tops[tid]; vi = tid; }
        block_argmax(v, vi, wv, wi, &selv_s, &seli_s);
        int i = seli_s; float sv = selv_s;
        bool ok = (i >= 0);
        if (tid < 5) {
            float val = 0.0f;
            if (ok) val = (tid == 0) ? sv : cb[i][tid - 1];
            orow[step * 5 + tid] = val;
        }
        if (ok && tid < 7) {
            unsigned m = msk[i][tid];
            if (tid == (i >> 5)) m &= ~(1u << (i & 31));
            aw[tid] &= m;
        }
        __syncthreads();
    }
}

// ============ Kernel D: zero the background-class block ============
__global__ __launch_bounds__(256)
void zero_bg_kernel(float* __restrict__ out) {
    int idx = blockIdx.x * blockDim.x + threadIdx.x;
    if (idx < B_ * TOPK * 5) {
        int b = idx / (TOPK * 5), r = idx % (TOPK * 5);
        out[(size_t)b * C_ * TOPK * 5 + r] = 0.0f;
    }
}

extern "C" void kernel_launch(void* const* d_in, const int* in_sizes, int n_in,
                              void* d_out, int out_size, void* d_ws, size_t ws_size,
                              hipStream_t stream) {
    const float* loc  = (const float*)d_in[0];
    const float* conf = (const float*)d_in[1];
    const float* dbox = (const float*)d_in[2];
    float* out = (float*)d_out;

    float2* denom = (float2*)d_ws;                                             // 2.24 MB
    float4* boxes = (float4*)((char*)d_ws + (size_t)B_ * N_ * sizeof(float2)); // 4.47 MB

    int total = B_ * N_;
    softmax_decode_kernel<<<(total + 255) / 256, 256, 0, stream>>>(loc, conf, dbox, denom, boxes);
    nms_kernel<<<B_ * 80, 256, 0, stream>>>(conf, denom, boxes, out);
    zero_bg_kernel<<<(B_ * TOPK * 5 + 255) / 256, 256, 0, stream>>>(out);
}